// Attention_3882650436473
// MI455X (gfx1250) — compile-verified
//
#include <hip/hip_runtime.h>
#include <hip/hip_bf16.h>
#include <math.h>

#define RNN_SIZE 2048
#define ATT_HID  512
#define BATCH    64
#define ATT_SIZE 1024

typedef __attribute__((ext_vector_type(2))) float v2f;
typedef __attribute__((ext_vector_type(8))) float v8f;

// ---------------------------------------------------------------------------
// Kernel 1: att_h = h @ W_h2att^T + b_h2att   (M=64, N=512, K=2048, f32)
// One wave (32 lanes) per 16x16 output tile, V_WMMA_F32_16X16X4_F32, K step 4.
//
// f32 A-matrix 16x4 layout (ISA 7.12.2): lane L holds row M = L%16;
//   VGPR v holds K = (L/16)*2 + v.
// f32 B-matrix 4x16 layout: lane L holds col N = L%16;
//   VGPR v holds K = (L/16)*2 + v.
// Since B[k][n] = W_h2att[n][k] (row-major W), both operands load float2 from
// the row (L%16) of their source at column (kbase + (L/16)*2).
//
// f32 C/D 16x16 layout: VGPR r, lane L -> element (M = r + 8*(L/16), N = L%16).
// ---------------------------------------------------------------------------
__global__ __launch_bounds__(128)
void att_h2att_wmma_kernel(const float* __restrict__ h,
                           const float* __restrict__ W_h2att,
                           const float* __restrict__ b_h2att,
                           float* __restrict__ att_h) {
    const int lane   = threadIdx.x & 31;
    const int waveId = (blockIdx.x * (blockDim.x >> 5)) + (threadIdx.x >> 5);
    const int mt = waveId >> 5;   // 0..3   (4 M-tiles of 16 over B=64)
    const int nt = waveId & 31;   // 0..31  (32 N-tiles of 16 over ATT_HID=512)

    const int row   = lane & 15;        // A row / B col within tile
    const int khalf = (lane >> 4) * 2;  // 0 or 2

    const float* __restrict__ aRow = h       + (size_t)(mt * 16 + row) * RNN_SIZE;
    const float* __restrict__ bRow = W_h2att + (size_t)(nt * 16 + row) * RNN_SIZE;

    v8f acc = {};
    #pragma unroll 4
    for (int k = 0; k < RNN_SIZE; k += 4) {
        const float2 af = *(const float2*)(aRow + k + khalf);
        const float2 bf = *(const float2*)(bRow + k + khalf);
        v2f a; a.x = af.x; a.y = af.y;
        v2f b; b.x = bf.x; b.y = bf.y;
        // 8 args: (neg_a, A, neg_b, B, c_mod, C, reuse_a, reuse_b)
        acc = __builtin_amdgcn_wmma_f32_16x16x4_f32(
            false, a, false, b, (short)0, acc, false, false);
    }

    const int n    = nt * 16 + (lane & 15);
    const float bi = b_h2att[n];
    const int mEdge = mt * 16 + 8 * (lane >> 4);
    #pragma unroll
    for (int r = 0; r < 8; ++r) {
        att_h[(size_t)(mEdge + r) * ATT_HID + n] = acc[r] + bi;
    }
}

// ---------------------------------------------------------------------------
// Kernel 2: scores[b,s] = sum_h tanh(p_att_feats[b,s,h] + att_h[b,h]) * w_alpha[h]
//           + b_alpha
// One wave per (b,s) row; 8 waves (256 threads) per block. Streams 128 MB.
// ---------------------------------------------------------------------------
__global__ __launch_bounds__(256)
void att_scores_kernel(const float* __restrict__ p_att_feats,
                       const float* __restrict__ att_h,
                       const float* __restrict__ w_alpha,
                       const float* __restrict__ b_alpha,
                       float* __restrict__ scores) {
    const int b    = blockIdx.x;
    const int wave = threadIdx.x >> 5;
    const int lane = threadIdx.x & 31;
    const int s    = blockIdx.y * 8 + wave;

    const float* __restrict__ prow = p_att_feats
        + ((size_t)b * ATT_SIZE + s) * ATT_HID;
    const float* __restrict__ ah = att_h + (size_t)b * ATT_HID;

    float acc = 0.0f;
    #pragma unroll 4
    for (int i = lane; i < ATT_HID; i += 32) {
        acc += w_alpha[i] * tanhf(prow[i] + ah[i]);
    }
    // wave32 reduction
    #pragma unroll
    for (int off = 16; off > 0; off >>= 1)
        acc += __shfl_xor(acc, off, 32);

    if (lane == 0) scores[(size_t)b * ATT_SIZE + s] = acc + b_alpha[0];
}

// ---------------------------------------------------------------------------
// Kernel 3: softmax over s (1024) per batch row. One 256-thread block per b.
// ---------------------------------------------------------------------------
__global__ __launch_bounds__(256)
void att_softmax_kernel(const float* __restrict__ scores,
                        float* __restrict__ weight) {
    const int b    = blockIdx.x;
    const int t    = threadIdx.x;
    const int lane = t & 31;
    const int wv   = t >> 5;
    __shared__ float red[8];

    const float* __restrict__ srow = scores + (size_t)b * ATT_SIZE;
    float*       __restrict__ wrow = weight + (size_t)b * ATT_SIZE;

    float m = -3.402823466e+38f;
    for (int s = t; s < ATT_SIZE; s += 256) m = fmaxf(m, srow[s]);
    #pragma unroll
    for (int off = 16; off > 0; off >>= 1)
        m = fmaxf(m, __shfl_xor(m, off, 32));
    if (lane == 0) red[wv] = m;
    __syncthreads();
    m = red[0];
    #pragma unroll
    for (int i = 1; i < 8; ++i) m = fmaxf(m, red[i]);
    __syncthreads();

    float sum = 0.0f;
    for (int s = t; s < ATT_SIZE; s += 256) {
        const float e = expf(srow[s] - m);
        wrow[s] = e;
        sum += e;
    }
    #pragma unroll
    for (int off = 16; off > 0; off >>= 1)
        sum += __shfl_xor(sum, off, 32);
    if (lane == 0) red[wv] = sum;
    __syncthreads();
    sum = 0.0f;
    #pragma unroll
    for (int i = 0; i < 8; ++i) sum += red[i];
    const float inv = 1.0f / sum;

    for (int s = t; s < ATT_SIZE; s += 256) wrow[s] *= inv;
}

// ---------------------------------------------------------------------------
// Kernel 4: att_res[b,d] = sum_s weight[b,s] * att_feats[b,s,d]
// One thread per float4 of d (512 threads-of-float4 per b). Each wave load is
// 512 contiguous bytes (b128 per lane). weight[b,s] is wave-uniform -> scalar
// load. Streams the 512 MB tensor exactly once.
// ---------------------------------------------------------------------------
__global__ __launch_bounds__(256)
void att_res_kernel(const float* __restrict__ weight,
                    const float* __restrict__ att_feats,
                    float* __restrict__ out) {
    const int b  = blockIdx.y;
    const int d4 = blockIdx.x * blockDim.x + threadIdx.x;   // 0..511

    const float4* __restrict__ base =
        (const float4*)(att_feats + (size_t)b * ATT_SIZE * RNN_SIZE) + d4;
    const float* __restrict__ w = weight + (size_t)b * ATT_SIZE;

    float4 acc = make_float4(0.f, 0.f, 0.f, 0.f);
    #pragma unroll 4
    for (int s = 0; s < ATT_SIZE; ++s) {
        if (s + 8 < ATT_SIZE)
            __builtin_prefetch(base + (size_t)(s + 8) * (RNN_SIZE / 4), 0, 0);
        const float ws = w[s];
        const float4 v = base[(size_t)s * (RNN_SIZE / 4)];
        acc.x += ws * v.x;
        acc.y += ws * v.y;
        acc.z += ws * v.z;
        acc.w += ws * v.w;
    }
    ((float4*)(out + (size_t)b * RNN_SIZE))[d4] = acc;
}

// ---------------------------------------------------------------------------
extern "C" void kernel_launch(void* const* d_in, const int* in_sizes, int n_in,
                              void* d_out, int out_size, void* d_ws, size_t ws_size,
                              hipStream_t stream) {
    const float* h           = (const float*)d_in[0];  // [64, 2048]
    const float* att_feats   = (const float*)d_in[1];  // [64, 1024, 2048]
    const float* p_att_feats = (const float*)d_in[2];  // [64, 1024, 512]
    const float* W_h2att     = (const float*)d_in[3];  // [512, 2048]
    const float* b_h2att     = (const float*)d_in[4];  // [512]
    const float* w_alpha     = (const float*)d_in[5];  // [512]
    const float* b_alpha     = (const float*)d_in[6];  // [1]
    float* out = (float*)d_out;                        // [64, 2048]

    float* att_h  = (float*)d_ws;                      // 64*512
    float* scores = att_h + (size_t)BATCH * ATT_HID;   // 64*1024
    float* weight = scores + (size_t)BATCH * ATT_SIZE; // 64*1024

    // 1) att_h GEMM: 128 tiles of 16x16, one wave each -> 32 blocks x 4 waves
    att_h2att_wmma_kernel<<<dim3(32), dim3(128), 0, stream>>>(
        h, W_h2att, b_h2att, att_h);

    // 2) scores: grid (64 batches, 1024/8 s-chunks), 8 waves per block
    att_scores_kernel<<<dim3(BATCH, ATT_SIZE / 8), dim3(256), 0, stream>>>(
        p_att_feats, att_h, w_alpha, b_alpha, scores);

    // 3) softmax per batch row
    att_softmax_kernel<<<dim3(BATCH), dim3(256), 0, stream>>>(scores, weight);

    // 4) weighted sum over att positions
    att_res_kernel<<<dim3((RNN_SIZE / 4) / 256, BATCH), dim3(256), 0, stream>>>(
        weight, att_feats, out);
}